// GCN_6786048327784
// MI455X (gfx1250) — compile-verified
//
#include <hip/hip_runtime.h>
#include <hip/hip_bf16.h>

typedef float v2f __attribute__((ext_vector_type(2)));
typedef float v8f __attribute__((ext_vector_type(8)));
typedef unsigned int u32x4 __attribute__((ext_vector_type(4)));
typedef int          i32x4 __attribute__((ext_vector_type(4)));
typedef int          i32x8 __attribute__((ext_vector_type(8)));

#define N_NODES 100000
#define N_EDGES 1600000
#define F_INDIM 128
#define DEMB    128
#define LDIM    256
#define ODIM    2
#define NGRAPH  64

#define ATOMIC_ADD_F32(p, v) \
  __hip_atomic_fetch_add((p), (v), __ATOMIC_RELAXED, __HIP_MEMORY_SCOPE_AGENT)

// ---------------------------------------------------------------- utilities
__global__ void fill_kernel(float* __restrict__ p, float v, long n) {
    long i = (long)blockIdx.x * blockDim.x + threadIdx.x;
    if (i < n) p[i] = v;
}

// deg[c] starts at 1.0 (self loop weight); add edge weights by destination.
__global__ void deg_kernel(const int* __restrict__ col, const float* __restrict__ w,
                           float* __restrict__ deg, int E) {
    int e = blockIdx.x * blockDim.x + threadIdx.x;
    if (e < E) ATOMIC_ADD_F32(&deg[col[e]], w[e]);
}

__global__ void rsqrt_kernel(float* __restrict__ deg, int n) {
    int i = blockIdx.x * blockDim.x + threadIdx.x;
    if (i < n) {
        float d = deg[i];
        deg[i] = (d > 0.0f) ? rsqrtf(d) : 0.0f;
    }
}

// --------------------------------------------- TDM: 2-D f32 tile -> LDS
// Builds the Tensor DMA Descriptor (ISA ch.8) in uniform registers and issues
// tensor_load_to_lds. All fields wave-uniform -> SGPR groups as required.
__device__ __forceinline__ void tdm_load_f32_2d(unsigned lds_byte_addr,
                                                const void* gptr,
                                                unsigned dim0, unsigned dim1,
                                                unsigned stride0) {
    unsigned long long ga = (unsigned long long)gptr;
    u32x4 g0;
    g0.x = 1u;                                           // count=1, user D#
    g0.y = lds_byte_addr;                                // bits 63:32 lds_addr
    g0.z = (unsigned)(ga & 0xFFFFFFFFull);               // global_addr[31:0]
    g0.w = (unsigned)((ga >> 32) & 0x01FFFFFFull)        // global_addr[56:32]
         | (2u << 30);                                   // type = 2 ("image")
    i32x8 g1;
    g1[0] = (int)(2u << 16);                             // data_size=2 (4B)
    g1[1] = (int)((dim0 & 0xFFFFu) << 16);               // tensor_dim0[15:0]
    g1[2] = (int)(((dim0 >> 16) & 0xFFFFu)               // tensor_dim0[31:16]
         |  ((dim1 & 0xFFFFu) << 16));                   // tensor_dim1[15:0]
    g1[3] = (int)(((dim1 >> 16) & 0xFFFFu)               // tensor_dim1[31:16]
         |  ((dim0 & 0xFFFFu) << 16));                   // tile_dim0 = dim0
    g1[4] = (int)(dim1 & 0xFFFFu);                       // tile_dim1 = dim1
    g1[5] = (int)stride0;                                // tensor_dim0_stride
    g1[6] = 0;
    g1[7] = 0;
    i32x4 gz = {0, 0, 0, 0};                             // 2-D: groups 2/3 off
#if __clang_major__ >= 23
    i32x8 gz8 = {0, 0, 0, 0, 0, 0, 0, 0};
    __builtin_amdgcn_tensor_load_to_lds(g0, g1, gz, gz, gz8, 0);
#else
    __builtin_amdgcn_tensor_load_to_lds(g0, g1, gz, gz, 0);
#endif
}

// ---------------------------------------- conv GEMM: C[N,128] = A[N,128] @ B
// Fixed 128x128 f32 weight staged into LDS by the Tensor Data Mover; one wave
// per 16x16 output tile, fp32 WMMA 16x16x4, A rows shared across the block's
// 8 waves (WGP$-resident). Grid is an exact tile fit (no early return, so the
// barrier is safe and EXEC is all-ones at every WMMA).
__global__ void __launch_bounds__(256)
wmma_gemm_conv(const float* __restrict__ A, const float* __restrict__ B,
               float* __restrict__ C) {
    __shared__ float Bs[DEMB * DEMB];                    // 64 KB

    if (threadIdx.x < 32) {                              // wave 0 issues DMA
        unsigned lds_addr = (unsigned)(unsigned long long)
            (__attribute__((address_space(3))) float*)Bs;
        tdm_load_f32_2d(lds_addr, B, DEMB, DEMB, DEMB);
        __builtin_amdgcn_s_wait_tensorcnt(0);
    }
    __syncthreads();

    int wave = threadIdx.x >> 5;          // 0..7 -> n-tile
    int m0   = blockIdx.x << 4;
    int n0   = wave << 4;
    int lane = threadIdx.x & 31;
    int mr   = lane & 15;
    int half = lane >> 4;

    v8f acc = {0.f, 0.f, 0.f, 0.f, 0.f, 0.f, 0.f, 0.f};
    const float* arow = A + (long)(m0 + mr) * DEMB;
    const float* bcol = Bs + n0 + mr;

#pragma unroll 8
    for (int k = 0; k < DEMB; k += 4) {
        int ka = k + (half << 1);
        v2f a = *(const v2f*)(arow + ka);                // global, coalesced
        v2f b;                                           // LDS reads
        b.x = bcol[ka * DEMB];
        b.y = bcol[(ka + 1) * DEMB];
        acc = __builtin_amdgcn_wmma_f32_16x16x4_f32(
                  false, a, false, b, (short)0, acc, false, false);
    }

    int colg = n0 + mr;
#pragma unroll
    for (int v = 0; v < 8; ++v) {
        int rowg = m0 + v + (half << 3);
        C[(long)rowg * DEMB + colg] = acc[v];
    }
}

// --------------------------- generic fp32 WMMA GEMM (global-load B path)
// Used for the small MLP head where B (up to 256 KB) exceeds LDS staging.
__global__ void __launch_bounds__(256)
wmma_gemm_f32(const float* __restrict__ A, const float* __restrict__ B,
              float* __restrict__ C, const float* __restrict__ bias,
              int M, int K, int Nc, int relu) {
    int wave   = blockIdx.x * (blockDim.x >> 5) + (threadIdx.x >> 5);
    int tilesN = Nc >> 4;
    int total  = (M >> 4) * tilesN;
    if (wave >= total) return;

    int m0   = (wave / tilesN) << 4;
    int n0   = (wave % tilesN) << 4;
    int lane = threadIdx.x & 31;
    int mr   = lane & 15;
    int half = lane >> 4;

    v8f acc = {0.f, 0.f, 0.f, 0.f, 0.f, 0.f, 0.f, 0.f};
    const float* arow = A + (long)(m0 + mr) * K;
    const float* bcol = B + n0 + mr;

    for (int k = 0; k < K; k += 4) {
        int ka = k + (half << 1);
        v2f a = *(const v2f*)(arow + ka);
        v2f b;
        b.x = bcol[(long)ka * Nc];
        b.y = bcol[(long)(ka + 1) * Nc];
        acc = __builtin_amdgcn_wmma_f32_16x16x4_f32(
                  false, a, false, b, (short)0, acc, false, false);
    }

    int   colg = n0 + mr;
    float bv   = bias ? bias[colg] : 0.0f;
#pragma unroll
    for (int v = 0; v < 8; ++v) {
        int   rowg = m0 + v + (half << 3);
        float val  = acc[v] + bv;
        if (relu) val = fmaxf(val, 0.0f);
        C[(long)rowg * Nc + colg] = val;
    }
}

// ------------------------------------------------ gather * norm -> scatter
// One wave per (edge or self-loop); each lane handles 4 contiguous floats, so
// a wave's atomics cover one contiguous 512B destination row (L2-resident).
__global__ void aggregate_kernel(const float* __restrict__ h,
                                 const int* __restrict__ row,
                                 const int* __restrict__ col,
                                 const float* __restrict__ ew,
                                 const float* __restrict__ dinv,
                                 float* __restrict__ out, int E, int n) {
    long t   = (long)blockIdx.x * blockDim.x + threadIdx.x;
    long idx = t >> 5;
    int  f   = ((int)t & 31) << 2;
    if (idx >= (long)E + n) return;

    int r, c; float w;
    if (idx < E) { r = row[idx]; c = col[idx]; w = ew[idx]; }
    else         { r = c = (int)(idx - E);     w = 1.0f;    }

    float  norm = dinv[r] * w * dinv[c];
    float4 hv   = *(const float4*)(h + (long)r * DEMB + f);
    float* o    = out + (long)c * DEMB + f;
    ATOMIC_ADD_F32(o + 0, hv.x * norm);
    ATOMIC_ADD_F32(o + 1, hv.y * norm);
    ATOMIC_ADD_F32(o + 2, hv.z * norm);
    ATOMIC_ADD_F32(o + 3, hv.w * norm);
}

// dst = relu(agg + bias[d]); agg reset to 0 for the next conv's accumulation.
__global__ void biasrelu_reset_kernel(float* __restrict__ agg,
                                      const float* __restrict__ bias,
                                      float* __restrict__ dst, long n) {
    long i = (long)blockIdx.x * blockDim.x + threadIdx.x;
    if (i < n) {
        float v = agg[i] + bias[(int)(i & (DEMB - 1))];
        dst[i]  = fmaxf(v, 0.0f);
        agg[i]  = 0.0f;
    }
}

// Fused: relu(agg + b2) scattered into per-graph sums (mean-pool numerator).
__global__ void biasrelu_pool_kernel(const float* __restrict__ agg,
                                     const float* __restrict__ bias,
                                     const int* __restrict__ batch,
                                     float* __restrict__ sums, int n) {
    long t    = (long)blockIdx.x * blockDim.x + threadIdx.x;
    long node = t >> 5;
    int  f    = ((int)t & 31) << 2;
    if (node >= n) return;
    int    g = batch[node];
    float4 v = *(const float4*)(agg + node * DEMB + f);
    float* s = sums + (long)g * DEMB + f;
    ATOMIC_ADD_F32(s + 0, fmaxf(v.x + bias[f + 0], 0.0f));
    ATOMIC_ADD_F32(s + 1, fmaxf(v.y + bias[f + 1], 0.0f));
    ATOMIC_ADD_F32(s + 2, fmaxf(v.z + bias[f + 2], 0.0f));
    ATOMIC_ADD_F32(s + 3, fmaxf(v.w + bias[f + 3], 0.0f));
}

__global__ void count_kernel(const int* __restrict__ batch,
                             float* __restrict__ cnt, int n) {
    int i = blockIdx.x * blockDim.x + threadIdx.x;
    if (i < n) ATOMIC_ADD_F32(&cnt[batch[i]], 1.0f);
}

__global__ void mean_kernel(float* __restrict__ sums,
                            const float* __restrict__ cnt) {
    int i = blockIdx.x * blockDim.x + threadIdx.x;   // NGRAPH*DEMB threads
    if (i < NGRAPH * DEMB) sums[i] /= fmaxf(cnt[i >> 7], 1.0f);
}

// Final 256 -> 2 projection on 64 rows (tiny; scalar is fine).
__global__ void final_kernel(const float* __restrict__ g2,
                             const float* __restrict__ lw3,
                             const float* __restrict__ lb3,
                             float* __restrict__ out) {
    int t = blockIdx.x * blockDim.x + threadIdx.x;
    if (t >= NGRAPH * ODIM) return;
    int g = t / ODIM, o = t % ODIM;
    float s = lb3[o];
    for (int k = 0; k < LDIM; ++k) s += g2[g * LDIM + k] * lw3[k * ODIM + o];
    out[t] = s;
}

// ---------------------------------------------------------------- launcher
extern "C" void kernel_launch(void* const* d_in, const int* in_sizes, int n_in,
                              void* d_out, int out_size, void* d_ws, size_t ws_size,
                              hipStream_t stream) {
    const float* x   = (const float*)d_in[0];
    const float* ew  = (const float*)d_in[1];
    const int*   ei  = (const int*)  d_in[2];
    const int*   bat = (const int*)  d_in[3];
    const float* W1  = (const float*)d_in[4];
    const float* b1  = (const float*)d_in[5];
    const float* W2  = (const float*)d_in[6];
    const float* b2  = (const float*)d_in[7];
    const float* lw1 = (const float*)d_in[8];
    const float* lb1 = (const float*)d_in[9];
    const float* lw2 = (const float*)d_in[10];
    const float* lb2 = (const float*)d_in[11];
    const float* lw3 = (const float*)d_in[12];
    const float* lb3 = (const float*)d_in[13];
    float*       out = (float*)d_out;

    const int* row = ei;            // edge_index[0]
    const int* col = ei + N_EDGES;  // edge_index[1]

    // workspace carve-up (floats)
    const long NH = (long)N_NODES * DEMB;          // 12.8M
    float* ws   = (float*)d_ws;
    float* agg  = ws;                              // [N, 128] accumulator
    float* h    = ws + NH;                         // [N, 128] GEMM out
    float* h2   = ws + 2 * NH;                     // [N, 128] conv1 result
    float* deg  = ws + 3 * NH;                     // [N] degree -> d^{-1/2}
    float* sums = deg + N_NODES;                   // [64, 128] pool sums
    float* cnt  = sums + NGRAPH * DEMB;            // [64]
    float* g1   = cnt + NGRAPH;                    // [64, 256]
    float* g2   = g1 + (long)NGRAPH * LDIM;        // [64, 256]

    const int T = 256;
    const int convGrid = N_NODES / 16;                                    // 6250
    const long aggT    = ((long)N_EDGES + N_NODES) * 32;
    const int  aggGrid = (int)((aggT + T - 1) / T);

    // init
    fill_kernel<<<(int)((NH + T - 1) / T), T, 0, stream>>>(agg, 0.0f, NH);
    fill_kernel<<<(N_NODES + T - 1) / T, T, 0, stream>>>(deg, 1.0f, N_NODES);
    fill_kernel<<<(NGRAPH * DEMB + NGRAPH + T - 1) / T, T, 0, stream>>>(
        sums, 0.0f, NGRAPH * DEMB + NGRAPH);

    // symmetric degree norm
    deg_kernel<<<(N_EDGES + T - 1) / T, T, 0, stream>>>(col, ew, deg, N_EDGES);
    rsqrt_kernel<<<(N_NODES + T - 1) / T, T, 0, stream>>>(deg, N_NODES);

    // conv1: h = x @ W1 (TDM-staged W) ; aggregate ; h2 = relu(agg+b1), agg=0
    wmma_gemm_conv<<<convGrid, T, 0, stream>>>(x, W1, h);
    aggregate_kernel<<<aggGrid, T, 0, stream>>>(h, row, col, ew, deg, agg,
                                                N_EDGES, N_NODES);
    biasrelu_reset_kernel<<<(int)((NH + T - 1) / T), T, 0, stream>>>(agg, b1, h2, NH);

    // conv2: h = h2 @ W2 ; aggregate ; fused relu(agg+b2) + mean-pool scatter
    wmma_gemm_conv<<<convGrid, T, 0, stream>>>(h2, W2, h);
    aggregate_kernel<<<aggGrid, T, 0, stream>>>(h, row, col, ew, deg, agg,
                                                N_EDGES, N_NODES);
    biasrelu_pool_kernel<<<(int)(((long)N_NODES * 32 + T - 1) / T), T, 0, stream>>>(
        agg, b2, bat, sums, N_NODES);
    count_kernel<<<(N_NODES + T - 1) / T, T, 0, stream>>>(bat, cnt, N_NODES);
    mean_kernel<<<(NGRAPH * DEMB + T - 1) / T, T, 0, stream>>>(sums, cnt);

    // MLP head: two WMMA GEMMs (bias+relu fused) + tiny final projection
    wmma_gemm_f32<<<((NGRAPH / 16) * (LDIM / 16) + 7) / 8, T, 0, stream>>>(
        sums, lw1, g1, lb1, NGRAPH, DEMB, LDIM, 1);
    wmma_gemm_f32<<<((NGRAPH / 16) * (LDIM / 16) + 7) / 8, T, 0, stream>>>(
        g1, lw2, g2, lb2, NGRAPH, LDIM, LDIM, 1);
    final_kernel<<<1, 128, 0, stream>>>(g2, lw3, lb3, out);
}